// GCNLayer_47210280517996
// MI455X (gfx1250) — compile-verified
//
#include <hip/hip_runtime.h>
#include <cstdint>
#include <cstddef>

#define DFEAT 128
#define BN_EPS 1e-5f

typedef __attribute__((ext_vector_type(2))) float v2f;
typedef __attribute__((ext_vector_type(8))) float v8f;

// ---------------------------------------------------------------------------
// Kernel 1: zero the aggregation buffer + stat accumulators (float4 stores)
// ---------------------------------------------------------------------------
__global__ __launch_bounds__(256) void zero_kernel(float* __restrict__ p, long n) {
    long i4 = ((long)blockIdx.x * blockDim.x + threadIdx.x) * 4;
    if (i4 + 3 < n) {
        *(float4*)(p + i4) = make_float4(0.f, 0.f, 0.f, 0.f);
    } else {
        for (long j = i4; j < n; ++j) p[j] = 0.f;
    }
}

// ---------------------------------------------------------------------------
// Kernel 2: edge scatter-add.  One wave32 per edge; lane l handles features
// [4l, 4l+4).  Whole wave reads 512B contiguous of x[src], then issues
// 4 fp32 atomic adds per lane into aggr[dst].
// ---------------------------------------------------------------------------
__global__ __launch_bounds__(256) void scatter_kernel(
    const float* __restrict__ x, const int* __restrict__ src,
    const int* __restrict__ dst, const float* __restrict__ ew,
    float* __restrict__ aggr, int E)
{
    long gid = (long)blockIdx.x * blockDim.x + threadIdx.x;
    int e    = (int)(gid >> 5);
    int lane = (int)(gid & 31);
    if (e >= E) return;
    int s = src[e];
    int d = dst[e];
    float w = ew[e];
    float4 v = ((const float4*)(x + (size_t)s * DFEAT))[lane];
    float* ap = aggr + (size_t)d * DFEAT + lane * 4;
    unsafeAtomicAdd(ap + 0, v.x * w);
    unsafeAtomicAdd(ap + 1, v.y * w);
    unsafeAtomicAdd(ap + 2, v.z * w);
    unsafeAtomicAdd(ap + 3, v.w * w);
}

// ---------------------------------------------------------------------------
// Kernel 3: h = aggr @ W_rel + x @ W_root + b_rel  via V_WMMA_F32_16X16X4_F32.
// Block = 128 threads (4 waves), computes a 16-row x 128-col tile of h.
// Each wave owns two 16x16 N-tiles; K-loop steps by 4 (32 iters, 4 WMMA/iter).
// A-frag (16x4 f32): lanes 0-15 hold K=k0,k0+1; lanes 16-31 hold K=k0+2,k0+3.
// B-frag (4x16):     v.x = W[kbase][col], v.y = W[kbase+1][col].
// C/D (16x16 f32):   VGPR j: lanes 0-15 -> M=j, lanes 16-31 -> M=j+8.
// Also accumulates per-column sum / sum-of-squares for the BatchNorm.
// ---------------------------------------------------------------------------
__global__ __launch_bounds__(128) void gemm_stats_kernel(
    const float* __restrict__ aggr, const float* __restrict__ x,
    const float* __restrict__ Wrel, const float* __restrict__ Wroot,
    const float* __restrict__ brel,
    float* __restrict__ h, float* __restrict__ colsum, float* __restrict__ colsumsq)
{
    const int lane  = threadIdx.x & 31;
    const int wave  = threadIdx.x >> 5;
    const int m     = lane & 15;       // A row / B,C column within tile
    const int khalf = lane >> 4;       // 0 or 1 -> K sub-pair
    const int rowBase = blockIdx.x * 16;
    const size_t rowOff = (size_t)(rowBase + m) * DFEAT;
    const int col0 = wave * 32 + m;
    const int col1 = col0 + 16;

    v8f c0 = {};
    v8f c1 = {};

    for (int k0 = 0; k0 < DFEAT; k0 += 4) {
        const int ka = k0 + khalf * 2;
        v2f a  = *(const v2f*)(aggr + rowOff + ka);   // A-frag for lin_rel
        v2f ax = *(const v2f*)(x    + rowOff + ka);   // A-frag for lin_root
        v2f br0, br1, bo0, bo1;
        br0.x = Wrel [(size_t)ka * DFEAT + col0];  br0.y = Wrel [(size_t)(ka + 1) * DFEAT + col0];
        br1.x = Wrel [(size_t)ka * DFEAT + col1];  br1.y = Wrel [(size_t)(ka + 1) * DFEAT + col1];
        bo0.x = Wroot[(size_t)ka * DFEAT + col0];  bo0.y = Wroot[(size_t)(ka + 1) * DFEAT + col0];
        bo1.x = Wroot[(size_t)ka * DFEAT + col1];  bo1.y = Wroot[(size_t)(ka + 1) * DFEAT + col1];

        c0 = __builtin_amdgcn_wmma_f32_16x16x4_f32(false, a,  false, br0, (short)0, c0, false, false);
        c1 = __builtin_amdgcn_wmma_f32_16x16x4_f32(false, a,  false, br1, (short)0, c1, false, false);
        c0 = __builtin_amdgcn_wmma_f32_16x16x4_f32(false, ax, false, bo0, (short)0, c0, false, false);
        c1 = __builtin_amdgcn_wmma_f32_16x16x4_f32(false, ax, false, bo1, (short)0, c1, false, false);
    }

    const float bb0 = brel[col0];
    const float bb1 = brel[col1];
    const int mrowBase = rowBase + khalf * 8;

    float s0 = 0.f, q0 = 0.f, s1 = 0.f, q1 = 0.f;
    #pragma unroll
    for (int j = 0; j < 8; ++j) {
        float v0 = c0[j] + bb0;
        float v1 = c1[j] + bb1;
        h[(size_t)(mrowBase + j) * DFEAT + col0] = v0;
        h[(size_t)(mrowBase + j) * DFEAT + col1] = v1;
        s0 += v0;  q0 += v0 * v0;
        s1 += v1;  q1 += v1 * v1;
    }
    // lane L and L+16 hold the same columns: pair-reduce, lanes 0-15 commit.
    s0 += __shfl_xor(s0, 16, 32);
    q0 += __shfl_xor(q0, 16, 32);
    s1 += __shfl_xor(s1, 16, 32);
    q1 += __shfl_xor(q1, 16, 32);
    if (khalf == 0) {
        unsafeAtomicAdd(&colsum[col0],   s0);
        unsafeAtomicAdd(&colsumsq[col0], q0);
        unsafeAtomicAdd(&colsum[col1],   s1);
        unsafeAtomicAdd(&colsumsq[col1], q1);
    }
}

// ---------------------------------------------------------------------------
// Kernel 4: finalize BN statistics -> per-column scale/shift
// ---------------------------------------------------------------------------
__global__ __launch_bounds__(DFEAT) void bn_finalize_kernel(
    const float* __restrict__ colsum, const float* __restrict__ colsumsq,
    const float* __restrict__ gamma, const float* __restrict__ beta,
    float* __restrict__ scale, float* __restrict__ shift, float invN)
{
    int i = threadIdx.x;
    float mean = colsum[i] * invN;
    float var  = colsumsq[i] * invN - mean * mean;
    float sc   = rsqrtf(var + BN_EPS) * gamma[i];
    scale[i] = sc;
    shift[i] = beta[i] - mean * sc;
}

// ---------------------------------------------------------------------------
// Kernel 5: out = relu(h * scale + shift) + x  (in-place on d_out, float4)
// ---------------------------------------------------------------------------
__global__ __launch_bounds__(256) void bn_relu_res_kernel(
    float* __restrict__ out, const float* __restrict__ x,
    const float* __restrict__ scale, const float* __restrict__ shift, long n4)
{
    long i = (long)blockIdx.x * blockDim.x + threadIdx.x;
    if (i >= n4) return;
    long base = i * 4;
    int  col  = (int)(base & (DFEAT - 1));
    float4 hv = *(const float4*)(out + base);
    float4 xv = *(const float4*)(x + base);
    float4 sc = *(const float4*)(scale + col);
    float4 sh = *(const float4*)(shift + col);
    float4 r;
    r.x = fmaxf(hv.x * sc.x + sh.x, 0.f) + xv.x;
    r.y = fmaxf(hv.y * sc.y + sh.y, 0.f) + xv.y;
    r.z = fmaxf(hv.z * sc.z + sh.z, 0.f) + xv.z;
    r.w = fmaxf(hv.w * sc.w + sh.w, 0.f) + xv.w;
    *(float4*)(out + base) = r;
}

// ---------------------------------------------------------------------------
extern "C" void kernel_launch(void* const* d_in, const int* in_sizes, int n_in,
                              void* d_out, int out_size, void* d_ws, size_t ws_size,
                              hipStream_t stream) {
    const float* x     = (const float*)d_in[0];
    const int*   eidx  = (const int*)  d_in[1];
    const float* ew    = (const float*)d_in[2];
    const float* Wrel  = (const float*)d_in[3];
    const float* brel  = (const float*)d_in[4];
    const float* Wroot = (const float*)d_in[5];
    const float* gamma = (const float*)d_in[6];
    const float* beta  = (const float*)d_in[7];
    float* out = (float*)d_out;

    const int N = in_sizes[0] / DFEAT;
    const int E = in_sizes[2];
    const int* src = eidx;
    const int* dst = eidx + E;

    // workspace layout: aggr [N*D] | colsum [D] | colsumsq [D] | scale [D] | shift [D]
    float* aggr     = (float*)d_ws;
    float* colsum   = aggr + (size_t)N * DFEAT;
    float* colsumsq = colsum + DFEAT;
    float* scale    = colsumsq + DFEAT;
    float* shift    = scale + DFEAT;

    // 1. zero aggr + stat accumulators
    long zn = (long)N * DFEAT + 2 * DFEAT;
    long zthreads = (zn + 3) / 4;
    zero_kernel<<<(zthreads + 255) / 256, 256, 0, stream>>>(aggr, zn);

    // 2. edge scatter-add (one wave32 per edge)
    long sthreads = (long)E * 32;
    scatter_kernel<<<(sthreads + 255) / 256, 256, 0, stream>>>(x, src, dst, ew, aggr, E);

    // 3. dual GEMM + bias via WMMA, h -> d_out (temp), accumulate BN stats
    gemm_stats_kernel<<<N / 16, 128, 0, stream>>>(aggr, x, Wrel, Wroot, brel,
                                                  out, colsum, colsumsq);

    // 4. finalize BN stats
    bn_finalize_kernel<<<1, DFEAT, 0, stream>>>(colsum, colsumsq, gamma, beta,
                                                scale, shift, 1.0f / (float)N);

    // 5. BN apply + relu + residual, in place on d_out
    long n4 = (long)N * DFEAT / 4;
    bn_relu_res_kernel<<<(n4 + 255) / 256, 256, 0, stream>>>(out, x, scale, shift, n4);
}